// SGPSpatialEncoder_81200651698780
// MI455X (gfx1250) — compile-verified
//
#include <hip/hip_runtime.h>
#include <hip/hip_bf16.h>

typedef __attribute__((ext_vector_type(2))) float v2f;
typedef __attribute__((ext_vector_type(8))) float v8f;

#define DFEAT 128
#define OUTW  640   // (K+2)*D = 5*128

// ---------------------------------------------------------------------------
// Pass 1: initialize d_out: cols [0,128) = x, cols [128,640) = 0
// One block per node, 160 threads, 4 floats (float4) per thread.
__global__ void init_out_kernel(const float* __restrict__ x,
                                float* __restrict__ out, int n_nodes) {
    int n = blockIdx.x;
    if (n >= n_nodes) return;
    int t = threadIdx.x;                 // 0..159
    size_t o = (size_t)n * OUTW + (size_t)t * 4;
    if (t < 32) {
        float4 v = *(const float4*)(x + (size_t)n * DFEAT + (size_t)t * 4);
        *(float4*)(out + o) = v;
    } else {
        float4 z = {0.f, 0.f, 0.f, 0.f};
        *(float4*)(out + o) = z;
    }
}

// Zero deg[N] and gsum[128] in workspace.
__global__ void zero_ws_kernel(float* __restrict__ deg,
                               float* __restrict__ gsum, int n_nodes) {
    int i = blockIdx.x * blockDim.x + threadIdx.x;
    if (i < n_nodes) deg[i] = 0.f;
    if (i < DFEAT)   gsum[i] = 0.f;
}

// ---------------------------------------------------------------------------
// deg[col[e]] += ew[e]
__global__ void deg_kernel(const int* __restrict__ col,
                           const float* __restrict__ ew,
                           float* __restrict__ deg, int nedges) {
    int e = blockIdx.x * blockDim.x + threadIdx.x;
    if (e < nedges) atomicAdd(&deg[col[e]], ew[e]);
}

// deg -> rsqrt(deg) in place (0 where deg<=0)
__global__ void dinv_kernel(float* __restrict__ deg, int n_nodes) {
    int i = blockIdx.x * blockDim.x + threadIdx.x;
    if (i < n_nodes) {
        float d = deg[i];
        deg[i] = (d > 0.f) ? __frsqrt_rn(d) : 0.f;
    }
}

// w[e] = dinv[row]*ew*dinv[col]
__global__ void edgew_kernel(const int* __restrict__ row,
                             const int* __restrict__ col,
                             const float* __restrict__ ew,
                             const float* __restrict__ dinv,
                             float* __restrict__ w, int nedges) {
    int e = blockIdx.x * blockDim.x + threadIdx.x;
    if (e < nedges) w[e] = dinv[row[e]] * ew[e] * dinv[col[e]];
}

// ---------------------------------------------------------------------------
// Propagate one hop: dst[col[e]*640 + lane*4 ..] += w[e] * src[row[e]*stride + ..]
// One wave32 per edge; each lane handles 4 features (b128 load, 4 f32 atomics).
__global__ void prop_kernel(const float* __restrict__ src, int src_stride,
                            float* __restrict__ dst,   // pre-offset to hop column
                            const int* __restrict__ row,
                            const int* __restrict__ col,
                            const float* __restrict__ w, int nedges) {
    int gwave = (blockIdx.x * blockDim.x + threadIdx.x) >> 5;
    int lane  = threadIdx.x & 31;
    if (gwave >= nedges) return;
    int   r  = row[gwave];
    int   c  = col[gwave];
    float ww = w[gwave];
    float4 v = *(const float4*)(src + (size_t)r * src_stride + (size_t)lane * 4);
    float* d = dst + (size_t)c * OUTW + (size_t)lane * 4;
    atomicAdd(d + 0, v.x * ww);
    atomicAdd(d + 1, v.y * ww);
    atomicAdd(d + 2, v.z * ww);
    atomicAdd(d + 3, v.w * ww);
}

// ---------------------------------------------------------------------------
// Column sums of x via V_WMMA_F32_16X16X4_F32 (exact f32 MACs).
// A = ones(16x4): every lane holds {1,1}. B = 4 nodes x 16 features:
//   lanes 0-15 -> nodes n0+0,n0+1 ; lanes 16-31 -> nodes n0+2,n0+3
// Every row of C accumulates the same 16 column sums; lane L reads C[0]
// for feature tile*16 + (L&15). One wave per (feature-tile, node-partition).
// Main loop runs over full 4-node chunks only (wave-uniform trip count ->
// unconditional loads, EXEC stays all-ones); at most one partial tail chunk
// is handled branchlessly via clamped indices + zero masks.
__global__ void mean_wmma_kernel(const float* __restrict__ x,
                                 float* __restrict__ gsum, int n_nodes) {
    int gwave  = (blockIdx.x * blockDim.x + threadIdx.x) >> 5;
    int lane   = threadIdx.x & 31;
    int nwaves = (gridDim.x * blockDim.x) >> 5;
    int tile   = gwave & 7;          // which 16-feature tile
    int part   = gwave >> 3;         // node partition
    int nparts = nwaves >> 3;

    v2f a = {1.0f, 1.0f};
    v8f c = {};
    int feat = tile * 16 + (lane & 15);
    int koff = (lane >> 4) * 2;      // 0 for lanes 0-15, 2 for lanes 16-31
    const float* xf = x + feat;      // column base
    const size_t step = (size_t)nparts * 4 * DFEAT;

    int n0 = part * 4;
    const float* p = xf + (size_t)n0 * DFEAT + (size_t)koff * DFEAT;
    // full chunks: n0+3 < n_nodes is uniform across the wave
    for (; n0 + 3 < n_nodes; n0 += nparts * 4, p += step) {
        v2f b;
        b.x = p[0];
        b.y = p[DFEAT];
        c = __builtin_amdgcn_wmma_f32_16x16x4_f32(
                false, a, false, b, (short)0, c, false, false);
    }
    // tail: at most one partial chunk; branchless (keeps EXEC all-ones)
    if (n0 < n_nodes) {
        int na = n0 + koff;
        int nb = na + 1;
        int lim = n_nodes - 1;
        float ma = (na <= lim) ? 1.0f : 0.0f;
        float mb = (nb <= lim) ? 1.0f : 0.0f;
        int ia = (na <= lim) ? na : lim;
        int ib = (nb <= lim) ? nb : lim;
        v2f b;
        b.x = xf[(size_t)ia * DFEAT] * ma;
        b.y = xf[(size_t)ib * DFEAT] * mb;
        c = __builtin_amdgcn_wmma_f32_16x16x4_f32(
                false, a, false, b, (short)0, c, false, false);
    }
    if (lane < 16) atomicAdd(&gsum[tile * 16 + lane], c[0]);
}

// Broadcast mean to cols [512,640): one block (32 threads) per node.
__global__ void gbroad_kernel(const float* __restrict__ gsum,
                              float* __restrict__ out, int n_nodes, float invN) {
    int n = blockIdx.x;
    if (n >= n_nodes) return;
    int t = threadIdx.x;             // 0..31
    float4 g = *(const float4*)(gsum + (size_t)t * 4);
    float4 r = {g.x * invN, g.y * invN, g.z * invN, g.w * invN};
    *(float4*)(out + (size_t)n * OUTW + 512 + (size_t)t * 4) = r;
}

// ---------------------------------------------------------------------------
extern "C" void kernel_launch(void* const* d_in, const int* in_sizes, int n_in,
                              void* d_out, int out_size, void* d_ws, size_t ws_size,
                              hipStream_t stream) {
    const float* x  = (const float*)d_in[0];            // [N,128]
    const int*   ei = (const int*)d_in[1];              // [2,E] (int32; JAX demotes int64)
    const float* ew = (const float*)d_in[2];            // [E]

    const int N = in_sizes[0] / DFEAT;
    const int E = in_sizes[2];
    const int* row = ei;
    const int* col = ei + E;

    float* out  = (float*)d_out;
    float* deg  = (float*)d_ws;                          // [N]  (becomes dinv)
    float* w    = deg + N;                               // [E]
    float* gsum = w + E;                                 // [128]

    // init output (x copy + zeros) and workspace accumulators
    init_out_kernel<<<N, 160, 0, stream>>>(x, out, N);
    zero_ws_kernel<<<(N + 255) / 256, 256, 0, stream>>>(deg, gsum, N);

    // GCN normalization
    deg_kernel<<<(E + 255) / 256, 256, 0, stream>>>(col, ew, deg, E);
    dinv_kernel<<<(N + 255) / 256, 256, 0, stream>>>(deg, N);
    edgew_kernel<<<(E + 255) / 256, 256, 0, stream>>>(row, col, ew, deg, w, E);

    // K = 3 hops, accumulated directly into strided output columns
    const int edgeBlocks = (E + 7) / 8;                  // 8 waves (edges) per 256-thread block
    prop_kernel<<<edgeBlocks, 256, 0, stream>>>(x,         DFEAT, out + 128, row, col, w, E);
    prop_kernel<<<edgeBlocks, 256, 0, stream>>>(out + 128, OUTW,  out + 256, row, col, w, E);
    prop_kernel<<<edgeBlocks, 256, 0, stream>>>(out + 256, OUTW,  out + 384, row, col, w, E);

    // node mean via WMMA f32 16x16x4, then broadcast
    mean_wmma_kernel<<<64, 256, 0, stream>>>(x, gsum, N);
    gbroad_kernel<<<N, 32, 0, stream>>>(gsum, out, N, 1.0f / (float)N);
}